// LabelCorrelationAwareLoss_28922309771824
// MI455X (gfx1250) — compile-verified
//
#include <hip/hip_runtime.h>
#include <hip/hip_bf16.h>

typedef float v2f __attribute__((ext_vector_type(2)));
typedef float v8f __attribute__((ext_vector_type(8)));

#define C_DIM 4096

// ---------------------------------------------------------------------------
// Wave32 all-lane sum using the CDNA5 matrix pipe.
// A (16x4 f32): lane L (0-15): {v0,v1} = A[L][0..1]; lane L (16-31): A[L-16][2..3].
// Set v0 = q, v1 = 0, B = all-ones 4x16  =>  D[m][n] = q(m) + q(m+16)  (all n).
// D layout: VGPR r, lanes 0-15 -> M=r, lanes 16-31 -> M=r+8.  Summing the 8
// result VGPRs gives lanes 0-15: sum(rows 0-7), lanes 16-31: sum(rows 8-15);
// one xor-16 shuffle completes the 32-lane total (broadcast to every lane).
// EXEC must be all ones at every call site (no divergence).
// ---------------------------------------------------------------------------
__device__ __forceinline__ float wave_reduce_wmma(float q) {
#if __has_builtin(__builtin_amdgcn_wmma_f32_16x16x4_f32)
    v2f a; a[0] = q;    a[1] = 0.0f;
    v2f b; b[0] = 1.0f; b[1] = 1.0f;
    v8f c = {};
    v8f d = __builtin_amdgcn_wmma_f32_16x16x4_f32(
        /*neg_a=*/false, a, /*neg_b=*/false, b,
        /*c_mod=*/(short)0, c, /*reuse_a=*/false, /*reuse_b=*/false);
    float t = ((d[0] + d[1]) + (d[2] + d[3])) + ((d[4] + d[5]) + (d[6] + d[7]));
    t += __shfl_xor(t, 16, 32);
    return t;
#else
    // fallback: log2 shuffle tree (should not be taken on gfx1250)
    for (int ofs = 16; ofs > 0; ofs >>= 1) q += __shfl_xor(q, ofs, 32);
    return q;
#endif
}

// Per-element math: 3 transcendentals total (exp(-|x|), exp(|x|), log).
__device__ __forceinline__ void elem(float x, float y,
                                     float& bce, float& sneg, float& spos, float& n1) {
    float ax = __builtin_fabsf(x);
    float e  = __expf(-ax);              // exp(-|x|)
    float ei = __expf(ax);               // exp(|x|)
    bce += fmaxf(x, 0.0f) - x * y + __logf(1.0f + e);   // stable BCE term
    bool pos = (y != 0.0f);
    bool xp  = (x >= 0.0f);
    float expx  = xp ? ei : e;           // exp(x)
    float expmx = xp ? e  : ei;          // exp(-x)
    sneg += pos ? 0.0f : expx;
    spos += pos ? expmx : 0.0f;
    n1   += pos ? 1.0f : 0.0f;
}

// Pass 1: one 256-thread block per row.  Coalesced float4 (global_load_b128)
// streaming, per-lane accumulation, WMMA wave reduction, LDS cross-wave
// combine, then per-row {bce_sum, rowterm} written to workspace.
__global__ void __launch_bounds__(256)
lcal_row_pass(const float* __restrict__ yhat, const float* __restrict__ ylab,
              float2* __restrict__ row_out) {
    const int row = blockIdx.x;
    const int tid = threadIdx.x;
    const float4* __restrict__ x4 = (const float4*)(yhat + (size_t)row * C_DIM);
    const float4* __restrict__ y4 = (const float4*)(ylab + (size_t)row * C_DIM);

    float a_bce = 0.0f, a_sneg = 0.0f, a_spos = 0.0f, a_n1 = 0.0f;

    const int nvec = C_DIM >> 2;                 // 1024 float4 per row
    #pragma unroll
    for (int i = tid; i < nvec; i += 256) {
        float4 x = x4[i];
        float4 y = y4[i];
        elem(x.x, y.x, a_bce, a_sneg, a_spos, a_n1);
        elem(x.y, y.y, a_bce, a_sneg, a_spos, a_n1);
        elem(x.z, y.z, a_bce, a_sneg, a_spos, a_n1);
        elem(x.w, y.w, a_bce, a_sneg, a_spos, a_n1);
    }

    // wave32 reductions on the matrix pipe (all lanes active here)
    float w_bce  = wave_reduce_wmma(a_bce);
    float w_sneg = wave_reduce_wmma(a_sneg);
    float w_spos = wave_reduce_wmma(a_spos);
    float w_n1   = wave_reduce_wmma(a_n1);

    __shared__ float lds[8][4];
    const int wave = tid >> 5, lane = tid & 31;
    if (lane == 0) {
        lds[wave][0] = w_bce;  lds[wave][1] = w_sneg;
        lds[wave][2] = w_spos; lds[wave][3] = w_n1;
    }
    __syncthreads();

    if (tid == 0) {
        float b = 0.0f, sn = 0.0f, sp = 0.0f, n1 = 0.0f;
        #pragma unroll
        for (int w = 0; w < 8; ++w) {
            b  += lds[w][0]; sn += lds[w][1];
            sp += lds[w][2]; n1 += lds[w][3];
        }
        float n0    = (float)C_DIM - n1;
        float denom = n0 * n1;
        float rowterm = (denom > 0.0f) ? (sn * sp / denom) : 0.0f;
        row_out[row] = make_float2(b, rowterm);
    }
}

// Pass 2: single block reduces the 8192 per-row pairs and writes the scalar.
__global__ void __launch_bounds__(256)
lcal_final_pass(const float2* __restrict__ row_in, float* __restrict__ out,
                int nrows) {
    const int tid = threadIdx.x;
    float sb = 0.0f, sr = 0.0f;
    for (int i = tid; i < nrows; i += 256) {
        float2 v = row_in[i];
        sb += v.x;
        sr += v.y;
    }
    sb = wave_reduce_wmma(sb);
    sr = wave_reduce_wmma(sr);

    __shared__ float lds[8][2];
    const int wave = tid >> 5, lane = tid & 31;
    if (lane == 0) { lds[wave][0] = sb; lds[wave][1] = sr; }
    __syncthreads();

    if (tid == 0) {
        float b = 0.0f, r = 0.0f;
        #pragma unroll
        for (int w = 0; w < 8; ++w) { b += lds[w][0]; r += lds[w][1]; }
        double mean = (double)b / ((double)nrows * (double)C_DIM)
                    + (double)r / (double)nrows;
        out[0] = (float)mean;
    }
}

extern "C" void kernel_launch(void* const* d_in, const int* in_sizes, int n_in,
                              void* d_out, int out_size, void* d_ws, size_t ws_size,
                              hipStream_t stream) {
    const float* y_hat = (const float*)d_in[0];
    const float* y     = (const float*)d_in[1];
    float*  out     = (float*)d_out;
    float2* row_buf = (float2*)d_ws;             // 8192 * 8 B = 64 KB scratch

    const int nrows = in_sizes[0] / C_DIM;       // 8192

    lcal_row_pass<<<nrows, 256, 0, stream>>>(y_hat, y, row_buf);
    lcal_final_pass<<<1, 256, 0, stream>>>(row_buf, out, nrows);
}